// GAT_Transductive_47115791237824
// MI455X (gfx1250) — compile-verified
//
#include <hip/hip_runtime.h>
#include <hip/hip_bf16.h>
#include <math.h>

// ---------------------------------------------------------------------------
// GAT transductive, 2 layers.  N=4096, F=512, H1=8, D1=8, C=16.
// Strategy:
//   h1 = x @ W1            -> fp32 WMMA GEMM (v_wmma_f32_16x16x4_f32)
//   es/ed per node         -> tiny kernel
//   attention + aggregate  -> per-row two-pass masked softmax (sparse adj)
//   a1 = elu(...)          -> fused
//   h2 = a1 @ W2           -> same WMMA GEMM
//   layer-2 attention      -> same attention kernel, H=1, D=16, no ELU
// ---------------------------------------------------------------------------

#define GAT_ALPHA 0.2f

typedef float v2f __attribute__((ext_vector_type(2)));
typedef float v8f __attribute__((ext_vector_type(8)));

// ---------------------------------------------------------------------------
// WMMA fp32 GEMM:  C(MxN) = A(MxK) @ B(KxN)
// Requirements: M%64==0, N%16==0, K%32==0.  Block = 128 threads (4 waves).
// Each block computes a 64x16 tile; each wave one 16x16 tile.
// A staged in LDS with padded stride 33 (gcd(33,64)=1 -> conflict-free
// per-lane row reads).  B staged as 32x16.
// ---------------------------------------------------------------------------
#define GEMM_KT 32

__global__ __launch_bounds__(128)
void gemm_wmma_f32(const float* __restrict__ A, const float* __restrict__ B,
                   float* __restrict__ C, int M, int N, int K)
{
    __shared__ float As[64 * 33];
    __shared__ float Bs[GEMM_KT * 16];

    const int t    = threadIdx.x;        // 0..127
    const int wave = t >> 5;             // 0..3
    const int lane = t & 31;             // wave32
    const int half = lane >> 4;          // 0: lanes 0-15, 1: lanes 16-31
    const int l15  = lane & 15;

    const int rowBase = blockIdx.y * 64;
    const int colBase = blockIdx.x * 16;

    v8f acc = {0.f, 0.f, 0.f, 0.f, 0.f, 0.f, 0.f, 0.f};

    for (int k0 = 0; k0 < K; k0 += GEMM_KT) {
        // --- stage A block: 64 rows x 32 cols = 512 float4, 4 per thread ---
        #pragma unroll
        for (int i = 0; i < 4; ++i) {
            int q   = t + 128 * i;       // float4 index
            int row = q >> 3;            // 8 float4 per row
            int c4  = q & 7;
            float4 v = *reinterpret_cast<const float4*>(
                A + (size_t)(rowBase + row) * K + k0 + c4 * 4);
            float* dst = &As[row * 33 + c4 * 4];
            dst[0] = v.x; dst[1] = v.y; dst[2] = v.z; dst[3] = v.w;
        }
        // --- stage B block: 32 rows x 16 cols = 128 float4, 1 per thread ---
        {
            int row = t >> 2;
            int c4  = t & 3;
            float4 v = *reinterpret_cast<const float4*>(
                B + (size_t)(k0 + row) * N + colBase + c4 * 4);
            float* dst = &Bs[row * 16 + c4 * 4];
            dst[0] = v.x; dst[1] = v.y; dst[2] = v.z; dst[3] = v.w;
        }
        __syncthreads();

        // --- 8 WMMA steps of K=4 ---
        const int arow = wave * 16 + l15;
        #pragma unroll
        for (int kb = 0; kb < GEMM_KT; kb += 4) {
            // A 16x4 layout: lanes 0-15 hold {K0,K1}, lanes 16-31 hold {K2,K3}
            v2f a, b;
            a.x = As[arow * 33 + kb + 2 * half + 0];
            a.y = As[arow * 33 + kb + 2 * half + 1];
            // B 4x16 layout mirrors A: lane picks column l15
            b.x = Bs[(kb + 2 * half + 0) * 16 + l15];
            b.y = Bs[(kb + 2 * half + 1) * 16 + l15];
            acc = __builtin_amdgcn_wmma_f32_16x16x4_f32(
                false, a, false, b, (short)0, acc, false, false);
        }
        __syncthreads();
    }

    // C/D layout: VGPR r -> (M = half*8 + r, N = l15)
    const int orow = rowBase + wave * 16 + half * 8;
    const int ocol = colBase + l15;
    #pragma unroll
    for (int r = 0; r < 8; ++r)
        C[(size_t)(orow + r) * N + ocol] = acc[r];
}

// ---------------------------------------------------------------------------
// Per-node attention scores: es[n,h] = <h[n,h,:], a_src[h,:]>, same for ed.
// ---------------------------------------------------------------------------
__global__ void node_scores(const float* __restrict__ hfeat,
                            const float* __restrict__ a_src,
                            const float* __restrict__ a_dst,
                            float* __restrict__ es, float* __restrict__ ed,
                            int N, int H, int D)
{
    int idx = blockIdx.x * blockDim.x + threadIdx.x;   // over N*H
    if (idx >= N * H) return;
    int n = idx / H;
    int h = idx - n * H;
    const float* hp = hfeat + (size_t)n * H * D + h * D;
    float s = 0.f, d2 = 0.f;
    for (int d = 0; d < D; ++d) {
        float v = hp[d];
        s  += v * a_src[h * D + d];
        d2 += v * a_dst[h * D + d];
    }
    es[idx] = s;
    ed[idx] = d2;
}

// ---------------------------------------------------------------------------
// Wave (32-lane) reductions.
// ---------------------------------------------------------------------------
__device__ __forceinline__ float waveMax(float v) {
    #pragma unroll
    for (int o = 16; o > 0; o >>= 1) v = fmaxf(v, __shfl_xor(v, o));
    return v;
}
__device__ __forceinline__ float waveSum(float v) {
    #pragma unroll
    for (int o = 16; o > 0; o >>= 1) v += __shfl_xor(v, o);
    return v;
}

// ---------------------------------------------------------------------------
// Masked-softmax attention + aggregation for one node row per block.
//   out[i,h,:] = sum_j softmax_j(leaky(es[i,h]+ed[j,h]))*hfeat[j,h,:]
// Non-edges are exactly 0 (exp(-1e9 - max) underflows in fp32, matching ref).
// Two passes over the adjacency row (max, then exp/sum/acc). 256 thr = 8 waves.
// ---------------------------------------------------------------------------
template <int H, int D, bool ELU>
__global__ __launch_bounds__(256)
void gat_attn(const float* __restrict__ hfeat, const float* __restrict__ es,
              const float* __restrict__ ed, const unsigned char* __restrict__ adj,
              float* __restrict__ out, int N)
{
    const int i    = blockIdx.x;
    const int t    = threadIdx.x;          // 0..255
    const int lane = t & 31;
    const int wave = t >> 5;               // 0..7

    __shared__ float sRed[8][H * D];
    __shared__ float sMaxF[H];
    __shared__ float sSumF[H];

    float esi[H];
    #pragma unroll
    for (int h = 0; h < H; ++h) esi[h] = es[(size_t)i * H + h];

    const unsigned char* arow = adj + (size_t)i * N;

    // ---- pass 1: per-head row max over valid edges ----
    float mh[H];
    #pragma unroll
    for (int h = 0; h < H; ++h) mh[h] = -1e30f;

    for (int j = t; j < N; j += 256) {
        if (j + 256 < N) __builtin_prefetch(arow + j + 256, 0, 0);
        if (arow[j]) {
            #pragma unroll
            for (int h = 0; h < H; ++h) {
                float s = esi[h] + ed[(size_t)j * H + h];
                s = (s > 0.f) ? s : GAT_ALPHA * s;
                mh[h] = fmaxf(mh[h], s);
            }
        }
    }
    #pragma unroll
    for (int h = 0; h < H; ++h) mh[h] = waveMax(mh[h]);
    if (lane == 0) {
        #pragma unroll
        for (int h = 0; h < H; ++h) sRed[wave][h] = mh[h];
    }
    __syncthreads();
    if (t < H) {
        float m = -1e30f;
        #pragma unroll
        for (int w = 0; w < 8; ++w) m = fmaxf(m, sRed[w][t]);
        sMaxF[t] = m;
    }
    __syncthreads();
    #pragma unroll
    for (int h = 0; h < H; ++h) mh[h] = sMaxF[h];
    __syncthreads();   // sRed free for reuse

    // ---- pass 2: exp-sum + weighted feature accumulation ----
    float sumh[H];
    float acc[H * D];
    #pragma unroll
    for (int h = 0; h < H; ++h) sumh[h] = 0.f;
    #pragma unroll
    for (int k = 0; k < H * D; ++k) acc[k] = 0.f;

    for (int j = t; j < N; j += 256) {
        if (arow[j]) {
            const float* hj = hfeat + (size_t)j * H * D;
            #pragma unroll
            for (int h = 0; h < H; ++h) {
                float s = esi[h] + ed[(size_t)j * H + h];
                s = (s > 0.f) ? s : GAT_ALPHA * s;
                float p = __expf(s - mh[h]);
                sumh[h] += p;
                #pragma unroll
                for (int d = 0; d < D; ++d)
                    acc[h * D + d] += p * hj[h * D + d];
            }
        }
    }

    // reduce sums
    #pragma unroll
    for (int h = 0; h < H; ++h) sumh[h] = waveSum(sumh[h]);
    if (lane == 0) {
        #pragma unroll
        for (int h = 0; h < H; ++h) sRed[wave][h] = sumh[h];
    }
    __syncthreads();
    if (t < H) {
        float s = 0.f;
        #pragma unroll
        for (int w = 0; w < 8; ++w) s += sRed[w][t];
        sSumF[t] = s;
    }
    __syncthreads();

    // reduce accumulators
    #pragma unroll
    for (int k = 0; k < H * D; ++k) acc[k] = waveSum(acc[k]);
    if (lane == 0) {
        #pragma unroll
        for (int k = 0; k < H * D; ++k) sRed[wave][k] = acc[k];
    }
    __syncthreads();
    if (t < H * D) {
        float v = 0.f;
        #pragma unroll
        for (int w = 0; w < 8; ++w) v += sRed[w][t];
        v /= sSumF[t / D];
        if (ELU) v = (v > 0.f) ? v : (__expf(v) - 1.f);
        out[(size_t)i * H * D + t] = v;
    }
}

// ---------------------------------------------------------------------------
// Host-side launch
// ---------------------------------------------------------------------------
extern "C" void kernel_launch(void* const* d_in, const int* in_sizes, int n_in,
                              void* d_out, int out_size, void* d_ws, size_t ws_size,
                              hipStream_t stream)
{
    (void)in_sizes; (void)n_in; (void)out_size; (void)ws_size;

    const int N  = 4096;
    const int F  = 512;
    const int H1 = 8, D1 = 8;     // layer 1: 8 heads x 8 dims = 64
    const int HD1 = H1 * D1;
    const int C  = 16;            // layer 2: 1 head x 16 dims

    const float*         x    = (const float*)d_in[0];
    const unsigned char* adj  = (const unsigned char*)d_in[1];  // jnp bool
    const float*         W1   = (const float*)d_in[2];          // (512,64)
    const float*         a1s  = (const float*)d_in[3];          // (8,8)
    const float*         a1d  = (const float*)d_in[4];
    const float*         W2   = (const float*)d_in[5];          // (64,16)
    const float*         a2s  = (const float*)d_in[6];          // (1,16)
    const float*         a2d  = (const float*)d_in[7];
    float*               out  = (float*)d_out;                  // (4096,16)

    // workspace layout (floats)
    float* ws  = (float*)d_ws;
    float* h1  = ws;                         // N*64
    float* es1 = h1  + (size_t)N * HD1;      // N*8
    float* ed1 = es1 + (size_t)N * H1;       // N*8
    float* a1  = ed1 + (size_t)N * H1;       // N*64
    float* h2  = a1  + (size_t)N * HD1;      // N*16
    float* es2 = h2  + (size_t)N * C;        // N
    float* ed2 = es2 + (size_t)N;            // N

    // ---- layer 1 ----
    // h1 = x @ W1   (M=4096, K=512, N=64)
    gemm_wmma_f32<<<dim3(HD1 / 16, N / 64), 128, 0, stream>>>(x, W1, h1, N, HD1, F);
    // per-node scores
    node_scores<<<(N * H1 + 255) / 256, 256, 0, stream>>>(h1, a1s, a1d, es1, ed1, N, H1, D1);
    // attention + aggregate + ELU -> a1
    gat_attn<8, 8, true><<<N, 256, 0, stream>>>(h1, es1, ed1, adj, a1, N);

    // ---- layer 2 ----
    // h2 = a1 @ W2   (M=4096, K=64, N=16)
    gemm_wmma_f32<<<dim3(C / 16, N / 64), 128, 0, stream>>>(a1, W2, h2, N, C, HD1);
    node_scores<<<(N + 255) / 256, 256, 0, stream>>>(h2, a2s, a2d, es2, ed2, N, 1, C);
    // final attention -> output logits (no activation)
    gat_attn<1, 16, false><<<N, 256, 0, stream>>>(h2, es2, ed2, adj, out, N);
}